// DiffuserFracSelfAttention_4380866641977
// MI455X (gfx1250) — compile-verified
//
#include <hip/hip_runtime.h>
#include <math.h>

// ---------------------------------------------------------------------------
// DiffuserFracSelfAttention for MI455X (gfx1250, wave32, WMMA).
// Heavy math runs on V_WMMA_F32_16X16X4_F32 (f32 in / f32 acc): the binomial
// power series has intermediates ~rho^9 ~ 7e33 with heavy cancellation, so the
// f32 matrix pipe is the correct precision choice (reference is fp32).
// Data movement: double-buffered LDS fed by GLOBAL_LOAD_ASYNC_TO_LDS_B128
// (ASYNCcnt / s_wait_asynccnt) when the toolchain exposes the builtins.
// ---------------------------------------------------------------------------

typedef __attribute__((ext_vector_type(2))) float v2f;
typedef __attribute__((ext_vector_type(8))) float v8f;

#define NNODES 2048
#define HSZ    768
#define BM 128
#define BN 64
#define BK 32
#define TPB 256   // 8 wave32s: 4 along M x 2 along N, each wave owns 32x32

#define A_PAD 36  // A row stride (floats): lane-halves hit disjoint bank sets
#define B_PAD 72  // B row stride (floats): k+2 half lands 16 banks away

#if __has_builtin(__builtin_amdgcn_global_load_async_to_lds_b128) && \
    __has_builtin(__builtin_amdgcn_s_wait_asynccnt)
#define HAVE_ASYNC_LDS 1
#else
#define HAVE_ASYNC_LDS 0
#endif

#if HAVE_ASYNC_LDS
// Builtin signature (from hipcc diagnostic): param0 is a pointer to a 16-byte
// int vector in the global address space; param1 the LDS-side counterpart.
typedef int vi4 __attribute__((vector_size(4 * sizeof(int))));

__device__ __forceinline__ void cp_async16(float* lds, const float* g) {
  __builtin_amdgcn_global_load_async_to_lds_b128(
      (__attribute__((address_space(1))) vi4*)const_cast<float*>(g),
      (__attribute__((address_space(3))) vi4*)lds,
      0, 0);
}
#endif

// ---------------------------------------------------------------------------
// f32 WMMA GEMM: C[M,N] = A[M,K] @ B[K,N]  (row-major, M%128==0, N%64==0, K%32==0)
// Optional epilogue: L += coefBase * rho^(-iiPow) * C  (power-series accumulate)
// ---------------------------------------------------------------------------
__global__ __launch_bounds__(TPB) void gemm_wmma_f32(
    const float* __restrict__ A, const float* __restrict__ B,
    float* __restrict__ C, int M, int K, int N,
    float* __restrict__ Lacc, const float* __restrict__ rhoPtr,
    float coefBase, int iiPow)
{
  __shared__ float As[2][BM][A_PAD];
  __shared__ float Bs[2][BK][B_PAD];

  const int tid  = threadIdx.x;
  const int lane = tid & 31;        // wave32
  const int wv   = tid >> 5;        // 0..7
  const int wr   = wv >> 1;         // 0..3  -> 32-row strip
  const int wc   = wv & 1;          // 0..1  -> 32-col strip
  const int l15  = lane & 15;
  const int lh   = lane >> 4;       // lane half
  const int koff = lh << 1;         // ISA f32 A/B layout: half h covers K=2h,2h+1

  const int tileM = blockIdx.y * BM;
  const int tileN = blockIdx.x * BN;

  // stage one 128x32 A tile + 32x64 B tile into LDS buffer `buf`
  auto stage = [&](int kb, int buf) {
#pragma unroll
    for (int it = 0; it < 4; ++it) {          // A: 1024 16B chunks
      const int c  = tid + it * TPB;
      const int ar = c >> 3, ac = (c & 7) << 2;
      const float* g = &A[(size_t)(tileM + ar) * K + (size_t)kb * BK + ac];
      float* l = &As[buf][ar][ac];
#if HAVE_ASYNC_LDS
      cp_async16(l, g);
#else
      *reinterpret_cast<float4*>(l) = *reinterpret_cast<const float4*>(g);
#endif
    }
#pragma unroll
    for (int it = 0; it < 2; ++it) {          // B: 512 16B chunks
      const int c  = tid + it * TPB;
      const int br = c >> 4, bc = (c & 15) << 2;
      const float* g = &B[((size_t)kb * BK + br) * N + tileN + bc];
      float* l = &Bs[buf][br][bc];
#if HAVE_ASYNC_LDS
      cp_async16(l, g);
#else
      *reinterpret_cast<float4*>(l) = *reinterpret_cast<const float4*>(g);
#endif
    }
  };

  v8f acc[2][2] = {{{}, {}}, {{}, {}}};

  const int ktiles = K / BK;
  stage(0, 0);

  const int m0 = wr * 32 + l15;         // A-frag rows (two 16-row frags)
  const int n0 = wc * 32 + l15;         // B-frag cols (two 16-col frags)

  for (int kb = 0; kb < ktiles; ++kb) {
    const int cur = kb & 1;
    const bool more = (kb + 1 < ktiles);
    if (more) stage(kb + 1, cur ^ 1);
#if HAVE_ASYNC_LDS
    // in-order completion: <=6 outstanding (next tile) => current tile landed
    __builtin_amdgcn_s_wait_asynccnt(6);
#endif
    __syncthreads();

#pragma unroll
    for (int k0 = 0; k0 < BK; k0 += 4) {
      v2f a0, a1, b0, b1;
      a0.x = As[cur][m0][k0 + koff];       a0.y = As[cur][m0][k0 + koff + 1];
      a1.x = As[cur][m0 + 16][k0 + koff];  a1.y = As[cur][m0 + 16][k0 + koff + 1];
      b0.x = Bs[cur][k0 + koff][n0];       b0.y = Bs[cur][k0 + koff + 1][n0];
      b1.x = Bs[cur][k0 + koff][n0 + 16];  b1.y = Bs[cur][k0 + koff + 1][n0 + 16];
      acc[0][0] = __builtin_amdgcn_wmma_f32_16x16x4_f32(false, a0, false, b0, (short)0, acc[0][0], false, false);
      acc[0][1] = __builtin_amdgcn_wmma_f32_16x16x4_f32(false, a0, false, b1, (short)0, acc[0][1], false, false);
      acc[1][0] = __builtin_amdgcn_wmma_f32_16x16x4_f32(false, a1, false, b0, (short)0, acc[1][0], false, false);
      acc[1][1] = __builtin_amdgcn_wmma_f32_16x16x4_f32(false, a1, false, b1, (short)0, acc[1][1], false, false);
    }
    __syncthreads();
  }
#if HAVE_ASYNC_LDS
  __builtin_amdgcn_s_wait_asynccnt(0);
#endif

  float coef = 0.0f;
  if (Lacc) {
    const float rho = *rhoPtr;
    coef = coefBase * __powf(rho, -(float)iiPow);  // (num/den)*(-1/rho)^ii
  }

  // C/D layout: VGPR r -> row (r + 8*laneHalf), col = lane&15
#pragma unroll
  for (int f = 0; f < 2; ++f) {
    const int mbase = tileM + wr * 32 + f * 16 + lh * 8;
#pragma unroll
    for (int g = 0; g < 2; ++g) {
      const int n = tileN + wc * 32 + g * 16 + l15;
#pragma unroll
      for (int r = 0; r < 8; ++r) {
        const size_t idx = (size_t)(mbase + r) * N + n;
        C[idx] = acc[f][g][r];
        if (Lacc) Lacc[idx] += coef * acc[f][g][r];
      }
    }
  }
}

// ---------------------------------------------------------------------------
// Per-row stats: s1_i = sum_j exp(adj_ij)        (rowsum of W)
//                s2_i = sum_j exp(exp(adj_ij))   (rowsum of exp(W), for rho)
// ---------------------------------------------------------------------------
__global__ __launch_bounds__(256) void rowstats_kernel(
    const float* __restrict__ adj, float* __restrict__ s1,
    float* __restrict__ s2, int N)
{
  __shared__ float r1[256], r2[256];
  const int row = blockIdx.x;
  float a1 = 0.0f, a2 = 0.0f;
  for (int j = threadIdx.x; j < N; j += 256) {
    const float w = expf(adj[(size_t)row * N + j]);
    a1 += w;
    a2 += expf(w);
  }
  r1[threadIdx.x] = a1; r2[threadIdx.x] = a2;
  __syncthreads();
  for (int s = 128; s > 0; s >>= 1) {
    if (threadIdx.x < s) {
      r1[threadIdx.x] += r1[threadIdx.x + s];
      r2[threadIdx.x] += r2[threadIdx.x + s];
    }
    __syncthreads();
  }
  if (threadIdx.x == 0) { s1[row] = r1[0]; s2[row] = r2[0]; }
}

__global__ __launch_bounds__(256) void rowmax_kernel(
    const float* __restrict__ s2, float* __restrict__ rho, int N)
{
  __shared__ float red[256];
  float m = -3.4e38f;
  for (int i = threadIdx.x; i < N; i += 256) m = fmaxf(m, s2[i]);
  red[threadIdx.x] = m;
  __syncthreads();
  for (int s = 128; s > 0; s >>= 1) {
    if (threadIdx.x < s) red[threadIdx.x] = fmaxf(red[threadIdx.x], red[threadIdx.x + s]);
    __syncthreads();
  }
  if (threadIdx.x == 0) *rho = red[0];
}

// Bmat = rho*I - exp(adj)/s1[i];  Bp = I;  L = rho*I
__global__ __launch_bounds__(256) void build_bmat_kernel(
    const float* __restrict__ adj, const float* __restrict__ s1,
    const float* __restrict__ rhoPtr, float* __restrict__ Bmat,
    float* __restrict__ Bp, float* __restrict__ L, int N)
{
  const int j = blockIdx.x * blockDim.x + threadIdx.x;
  const int i = blockIdx.y;
  const float rho = *rhoPtr;
  const size_t idx = (size_t)i * N + j;
  const float d = (i == j) ? 1.0f : 0.0f;
  Bmat[idx] = rho * d - expf(adj[idx]) / s1[i];
  Bp[idx]   = d;
  L[idx]    = rho * d;
}

__global__ __launch_bounds__(256) void diag_kernel(
    const float* __restrict__ L, float* __restrict__ dvec, int N)
{
  const int i = blockIdx.x * blockDim.x + threadIdx.x;
  if (i < N) dvec[i] = L[(size_t)i * N + i];
}

// In-place: L <- I - L/diag(L)[:,None]   (rho^gamma scaling cancels here)
__global__ __launch_bounds__(256) void build_m_kernel(
    float* __restrict__ L, const float* __restrict__ dvec, int N)
{
  const int j = blockIdx.x * blockDim.x + threadIdx.x;
  const int i = blockIdx.y;
  const size_t idx = (size_t)i * N + j;
  const float d = (i == j) ? 1.0f : 0.0f;
  L[idx] = d - L[idx] / dvec[i];
}

// WvT[k][e] = Wv[e][k]  (so v = hidden @ WvT via the row-major GEMM)
__global__ __launch_bounds__(256) void transpose_kernel(
    const float* __restrict__ Win, float* __restrict__ Wout, int n)
{
  const int e = blockIdx.x * blockDim.x + threadIdx.x;  // output col
  const int k = blockIdx.y;                             // output row
  Wout[(size_t)k * n + e] = Win[(size_t)e * n + k];
}

__global__ __launch_bounds__(256) void bias_kernel(
    float* __restrict__ v, const float* __restrict__ bv, int E)
{
  const int j = blockIdx.x * blockDim.x + threadIdx.x;
  const int i = blockIdx.y;
  if (j < E) v[(size_t)i * E + j] += bv[j];
}

// ---------------------------------------------------------------------------
extern "C" void kernel_launch(void* const* d_in, const int* in_sizes, int n_in,
                              void* d_out, int out_size, void* d_ws, size_t ws_size,
                              hipStream_t stream) {
  (void)in_sizes; (void)n_in; (void)out_size; (void)ws_size;
  const int N = NNODES, E = HSZ;
  const size_t NN = (size_t)N * N;

  const float* hidden = (const float*)d_in[0];
  // d_in[1] mask, d_in[3] Wq, d_in[4] bq, d_in[5] Wk, d_in[6] bk: dead branch
  const float* adj = (const float*)d_in[2];
  const float* Wv  = (const float*)d_in[7];
  const float* bv  = (const float*)d_in[8];
  float* out = (float*)d_out;

  float* w = (float*)d_ws;
  float* Bmat = w;                 // 16 MB
  float* BpA  = Bmat + NN;         // 16 MB (also h ping after series)
  float* BpB  = BpA + NN;          // 16 MB (also h pong after series)
  float* Lm   = BpB + NN;          // 16 MB, becomes M in place
  float* WvT  = Lm + NN;           // 768*768
  float* s1   = WvT + (size_t)E * E;
  float* s2   = s1 + N;
  float* rho  = s2 + N;
  float* dvec = rho + 1;

  // 1) row stats + rho
  rowstats_kernel<<<N, 256, 0, stream>>>(adj, s1, s2, N);
  rowmax_kernel<<<1, 256, 0, stream>>>(s2, rho, N);

  // 2) Bmat / Bp / L init
  dim3 g2(N / 256, N);
  build_bmat_kernel<<<g2, 256, 0, stream>>>(adj, s1, rho, Bmat, BpA, Lm, N);

  // 3) truncated binomial power series: Bp <- Bp@Bmat; L += coef_ii * Bp
  dim3 gS(N / BN, N / BM);   // 32 x 16
  double num = 1.0, den = 1.0;
  float* Acur = BpA; float* Cnext = BpB;
  for (int ii = 1; ii < 10; ++ii) {
    num *= (0.5 - ii + 1);
    den *= ii;
    const float coefBase = (float)((num / den) * ((ii & 1) ? -1.0 : 1.0));
    gemm_wmma_f32<<<gS, TPB, 0, stream>>>(Acur, Bmat, Cnext, N, N, N,
                                          Lm, rho, coefBase, ii);
    float* t = Acur; Acur = Cnext; Cnext = t;
  }

  // 4) M = I - L/diag(L)   (in place in Lm)
  diag_kernel<<<(N + 255) / 256, 256, 0, stream>>>(Lm, dvec, N);
  build_m_kernel<<<g2, 256, 0, stream>>>(Lm, dvec, N);

  // 5) v = hidden @ Wv.T + bv   (into BpA, now free)
  dim3 gT(E / 256, E);
  transpose_kernel<<<gT, 256, 0, stream>>>(Wv, WvT, E);
  dim3 gV(E / BN, N / BM);   // 12 x 16
  gemm_wmma_f32<<<gV, TPB, 0, stream>>>(hidden, WvT, BpA, N, E, E,
                                        nullptr, nullptr, 0.0f, 0);
  dim3 gB(E / 256, N);
  bias_kernel<<<gB, 256, 0, stream>>>(BpA, bv, E);

  // 6) five diffusion steps h <- M@h; final step writes d_out
  gemm_wmma_f32<<<gV, TPB, 0, stream>>>(Lm, BpA, BpB, N, N, E, nullptr, nullptr, 0.0f, 0);
  gemm_wmma_f32<<<gV, TPB, 0, stream>>>(Lm, BpB, BpA, N, N, E, nullptr, nullptr, 0.0f, 0);
  gemm_wmma_f32<<<gV, TPB, 0, stream>>>(Lm, BpA, BpB, N, N, E, nullptr, nullptr, 0.0f, 0);
  gemm_wmma_f32<<<gV, TPB, 0, stream>>>(Lm, BpB, BpA, N, N, E, nullptr, nullptr, 0.0f, 0);
  gemm_wmma_f32<<<gV, TPB, 0, stream>>>(Lm, BpA, out, N, N, E, nullptr, nullptr, 0.0f, 0);
}